// NNCL_6871947673993
// MI455X (gfx1250) — compile-verified
//
#include <hip/hip_runtime.h>

// ---------------------------------------------------------------------------
// Masked Woodbury reconstruction for MI455X (gfx1250, wave32, WMMA + async).
//   S = A^T A  (A column-orthonormal => S ~ I, cond(G) ~ 1+1e-6)
//   rhs = m .* ((x.*m) @ S)
//   Solve G u = rhs via masked Richardson:  u <- u + (rhs - m.*(u@S))
//   4 iterations => error << f32 eps; GEMMs via v_wmma_f32_16x16x4_f32.
// S is stored pair-packed: SP[((k>>2)*2 + ((k>>1)&1))*512 + n*2 + (k&1)]
// so a B-fragment is ONE coalesced global_load_b64 per lane per WMMA.
// Staging / write-back use GLOBAL_{LOAD,STORE}_ASYNC_*_LDS_B128 (ASYNCcnt).
// ---------------------------------------------------------------------------

typedef __attribute__((ext_vector_type(2))) float v2f;
typedef __attribute__((ext_vector_type(8))) float v8f;
typedef int v4i_g __attribute__((vector_size(16)));   // matches builtin pointee

#define BATCH   2048
#define DIM     256
#define DOUT    320
#define LDPAD   260      // 256+4: ds_load_b64 A-frags hit 64 distinct banks
#define T_ITERS 4

#define AS_G_V4(p) ((__attribute__((address_space(1))) v4i_g*)(p))
#define AS_L_V4(p) ((__attribute__((address_space(3))) v4i_g*)(p))

__device__ __forceinline__ v8f wmma_f32x4(v2f a, v2f b, v8f c) {
  return __builtin_amdgcn_wmma_f32_16x16x4_f32(false, a, false, b, (short)0, c,
                                               false, false);
}

__device__ __forceinline__ void async_g2l_b128(const void* g, void* l) {
#if __has_builtin(__builtin_amdgcn_global_load_async_to_lds_b128)
  __builtin_amdgcn_global_load_async_to_lds_b128(AS_G_V4(g), AS_L_V4(l), 0, 0);
#else
  *(float4*)l = *(const float4*)g;               // synchronous fallback
#endif
}

__device__ __forceinline__ void async_l2g_b128(void* g, const void* l) {
#if __has_builtin(__builtin_amdgcn_global_store_async_from_lds_b128)
  __builtin_amdgcn_global_store_async_from_lds_b128(AS_G_V4(g), AS_L_V4(l), 0, 0);
#else
  *(float4*)g = *(const float4*)l;               // synchronous fallback
#endif
}

__device__ __forceinline__ void wait_async() {
#if __has_builtin(__builtin_amdgcn_s_wait_asynccnt)
  __builtin_amdgcn_s_wait_asynccnt(0);
#endif
}

// ---------------------------------------------------------------------------
// Kernel 1: SP = pack(A^T A).  One wave per 16x16 tile, K-loop over 320.
// ---------------------------------------------------------------------------
__global__ void __launch_bounds__(32)
nncl_gram_kernel(const float* __restrict__ A, float* __restrict__ SP) {
  const int lane   = threadIdx.x & 31;
  const int lane16 = lane & 15;
  const int half   = lane >> 4;
  const int j0 = blockIdx.x * 16;
  const int i0 = blockIdx.y * 16;

  v8f acc = {};
#pragma unroll 8
  for (int kk = 0; kk < DOUT; kk += 4) {
    const int k0 = kk + 2 * half;
    v2f a, b;
    a.x = A[(size_t)k0 * DIM + i0 + lane16];
    a.y = A[(size_t)(k0 + 1) * DIM + i0 + lane16];
    b.x = A[(size_t)k0 * DIM + j0 + lane16];
    b.y = A[(size_t)(k0 + 1) * DIM + j0 + lane16];
    acc = wmma_f32x4(a, b, acc);
  }
#pragma unroll
  for (int v = 0; v < 8; ++v) {
    const int i = i0 + v + 8 * half;          // S row index (k of later GEMMs)
    const int j = j0 + lane16;                // S col index (n)
    const int idx = (((i >> 2) << 1) | ((i >> 1) & 1)) * 512 + j * 2 + (i & 1);
    SP[idx] = acc[v];
  }
}

// ---------------------------------------------------------------------------
// Kernel 2: 16 samples per workgroup; 8 waves, wave w owns N-tiles {w, w+8}.
// ---------------------------------------------------------------------------
__global__ void __launch_bounds__(256)
nncl_solve_kernel(const float* __restrict__ x, const int* __restrict__ mask,
                  const float* __restrict__ SP, float* __restrict__ out) {
  extern __shared__ float smem[];
  float* XR   = smem;                    // raw x        (async-load dest)
  int*   MKI  = (int*)(XR + 16 * LDPAD); // raw mask     (async-load dest)
  float* XM   = (float*)MKI + 16 * LDPAD;// x.*m  -> reused as XREC in phase 3
  float* MR   = XM  + 16 * LDPAD;        // mask as float
  float* RHS  = MR  + 16 * LDPAD;
  float* U0   = RHS + 16 * LDPAD;
  float* U1   = U0  + 16 * LDPAD;
  float* s_sum = U1 + 16 * LDPAD;        // 16
  float* s_sq  = s_sum + 16;             // 16
  float* s_ne  = s_sq  + 16;             // 16
  float* s_mse = s_ne  + 16;             // 16

  const int t      = threadIdx.x;
  const int r0     = blockIdx.x * 16;
  const int lane   = t & 31;
  const int lane16 = lane & 15;
  const int half   = (lane >> 4) & 1;
  const int w      = t >> 5;

  // ---- Phase 0a: async-stage raw x and mask rows into LDS ----
#pragma unroll
  for (int j = 0; j < 4; ++j) {
    const int chunk = t + 256 * j;              // 1024 x 16B chunks
    const int row   = chunk >> 6;
    const int c4    = (chunk & 63) << 2;
    async_g2l_b128(&x[(size_t)(r0 + row) * DIM + c4],    &XR[row * LDPAD + c4]);
    async_g2l_b128(&mask[(size_t)(r0 + row) * DIM + c4], &MKI[row * LDPAD + c4]);
  }
  if (t < 16) { s_sum[t] = 0.f; s_sq[t] = 0.f; s_ne[t] = 0.f; s_mse[t] = 0.f; }
  wait_async();
  __syncthreads();

  // ---- Phase 0b: XM = x.*m, MR = m, row stats via ds_add_f32 ----
#pragma unroll
  for (int i = 0; i < 16; ++i) {
    const float xv = XR[i * LDPAD + t];
    const float mv = MKI[i * LDPAD + t] ? 1.0f : 0.0f;
    XM[i * LDPAD + t] = xv * mv;
    MR[i * LDPAD + t] = mv;
    atomicAdd(&s_sum[i], xv);
    atomicAdd(&s_sq[i],  xv * xv);
    atomicAdd(&s_ne[i],  mv);
  }
  __syncthreads();

  // ---- Phase 1: RHS = m .* (XM @ S) ----
#pragma unroll
  for (int rep = 0; rep < 2; ++rep) {
    const int col0 = (w + 8 * rep) * 16;
    const int cn   = col0 + lane16;
    v8f acc = {};
#pragma unroll 8
    for (int kk = 0; kk < DIM; kk += 4) {
      const int k0 = kk + 2 * half;
      v2f a = *(const v2f*)&XM[lane16 * LDPAD + k0];                   // ds_load_b64
      v2f b = *(const v2f*)&SP[((kk >> 2) * 2 + half) * 512 + cn * 2]; // b64, coalesced
      acc = wmma_f32x4(a, b, acc);
    }
#pragma unroll
    for (int v = 0; v < 8; ++v) {
      const int M = v + 8 * half;
      const float r = acc[v] * MR[M * LDPAD + cn];
      RHS[M * LDPAD + cn] = r;
      U0[M * LDPAD + cn]  = r;
    }
  }
  __syncthreads();

  // ---- Phase 2: Richardson  u <- u + (rhs - m.*(u@S)), double-buffered ----
  float* Ua = U0;
  float* Ub = U1;
  for (int it = 0; it < T_ITERS; ++it) {
#pragma unroll
    for (int rep = 0; rep < 2; ++rep) {
      const int col0 = (w + 8 * rep) * 16;
      const int cn   = col0 + lane16;
      v8f acc = {};
#pragma unroll 8
      for (int kk = 0; kk < DIM; kk += 4) {
        const int k0 = kk + 2 * half;
        v2f a = *(const v2f*)&Ua[lane16 * LDPAD + k0];
        v2f b = *(const v2f*)&SP[((kk >> 2) * 2 + half) * 512 + cn * 2];
        acc = wmma_f32x4(a, b, acc);
      }
#pragma unroll
      for (int v = 0; v < 8; ++v) {
        const int M = v + 8 * half;
        Ub[M * LDPAD + cn] =
            Ua[M * LDPAD + cn] + RHS[M * LDPAD + cn] - MR[M * LDPAD + cn] * acc[v];
      }
    }
    __syncthreads();
    float* tmp = Ua; Ua = Ub; Ub = tmp;        // final solution ends in Ua
  }

  // ---- Phase 3a: x_rec into LDS (reuse XM), masked SSE ----
  float* XREC = XM;
#pragma unroll
  for (int i = 0; i < 16; ++i) {
    const float xv = XR[i * LDPAD + t];
    const float mv = MR[i * LDPAD + t];
    const float ne = s_ne[i];
    const float u  = Ua[i * LDPAD + t];
    const bool allE = (ne == (float)DIM);       // all-erased rows keep x
    const float xr = (allE || mv == 0.0f) ? xv : u;
    XREC[i * LDPAD + t] = xr;
    const float d = xr - xv;
    atomicAdd(&s_mse[i], mv * d * d);
  }
  __syncthreads();

  // ---- Phase 3b: async-drain x_rec from LDS to global; emit row metrics ----
#pragma unroll
  for (int j = 0; j < 4; ++j) {
    const int chunk = t + 256 * j;
    const int row   = chunk >> 6;
    const int c4    = (chunk & 63) << 2;
    async_l2g_b128(&out[(size_t)(r0 + row) * DIM + c4], &XREC[row * LDPAD + c4]);
  }
  if (t < 16) {
    const int row = r0 + t;
    const float ne   = s_ne[t];
    const float mse  = s_mse[t] / fmaxf(ne, 1.0f);
    const float mean = s_sum[t] * (1.0f / DIM);
    const float var  = s_sq[t] * (1.0f / DIM) - mean * mean;
    const float rel  = sqrtf(mse + 1e-9f) / sqrtf(var + 1e-9f);
    const float fr   = fminf(fmaxf(1.0f - rel, 0.0f), 1.0f);
    float* tail = out + (size_t)BATCH * DIM;
    tail[row]             = mse;   // mse
    tail[BATCH + row]     = fr;    // fr_acc
    tail[2 * BATCH + row] = ne;    // num_erased (float)
  }
  wait_async();                                  // drain ASYNCcnt before endpgm
}

// ---------------------------------------------------------------------------
extern "C" void kernel_launch(void* const* d_in, const int* in_sizes, int n_in,
                              void* d_out, int out_size, void* d_ws, size_t ws_size,
                              hipStream_t stream) {
  (void)in_sizes; (void)n_in; (void)out_size; (void)ws_size;
  const float* x    = (const float*)d_in[0];
  const float* A    = (const float*)d_in[1];
  const int*   mask = (const int*)d_in[2];
  float* SP  = (float*)d_ws;        // 256*256*4 = 256 KB, fully overwritten
  float* out = (float*)d_out;

  nncl_gram_kernel<<<dim3(DIM / 16, DIM / 16), 32, 0, stream>>>(A, SP);

  const size_t lds_bytes = (size_t)(7 * 16 * LDPAD + 64) * sizeof(float); // ~117 KB
  nncl_solve_kernel<<<BATCH / 16, 256, lds_bytes, stream>>>(x, mask, SP, out);
}